// _A3TGCN__56478819942832
// MI455X (gfx1250) — compile-verified
//
#include <hip/hip_runtime.h>
#include <hip/hip_bf16.h>

// ---------------------------------------------------------------------------
// A3TGCN (H=None) on gfx1250.
// Algebraic reduction: with H=0 the r-gate is dead and each gate's GEMM
// factors into pre[n,t,k] = agg[n,t]*aG[k] + cG[k], where
//   aG = conv_w @ lin_w[:32],  cG = conv_b @ lin_w[:32] + lin_b.
// The per-gate (z, h) 2x32x32 fusions are the only real matmuls -> done once with
// V_WMMA_F32_16X16X4_F32 (exact f32). Everything else is a memory-bound
// edge scatter (L2-resident atomics) + a VALU/transcendental node epilogue.
// ---------------------------------------------------------------------------

#define PERIODS 12
#define HIDDEN  32

typedef __attribute__((ext_vector_type(2))) float v2f;
typedef __attribute__((ext_vector_type(8))) float v8f;

// ---------------------------------------------------------------------------
// Kernel 0a: init scratch. deg starts at 1.0 (self-loop weight), agg at 0.
// ---------------------------------------------------------------------------
__global__ void k_init(float* __restrict__ deg, float* __restrict__ agg, int N) {
    int i = blockIdx.x * blockDim.x + threadIdx.x;
    if (i < N * PERIODS) agg[i] = 0.0f;
    if (i < N)           deg[i] = 1.0f;   // self-loop contributes 1.0 to in-degree
}

// ---------------------------------------------------------------------------
// Kernel 0b: fuse gate weights with WMMA (1 wave).
//   D[16x16] = A[16x4] @ B[4x16] accumulated over 8 K-chunks (K=32 total).
//   A row0 = conv_w, row1 = conv_b, rows 2..15 = 0.
//   Row0 of D -> aG, row1 of D (+lin_b) -> cG.
// A layout (ISA 7.12.2): lane L<16 holds A[L][k0], A[L][k0+1] with k0=kc*4;
// lane 16+L holds A[L][kc*4+2], A[L][kc*4+3]. B mirrors (rows across lanes).
// D layout: VGPR v, lanes 0-15 = D[v][n], lanes 16-31 = D[8+v][n].
// ---------------------------------------------------------------------------
__device__ __forceinline__ void gate_fuse(int lane,
                                          const float* __restrict__ cw,
                                          const float* __restrict__ cb,
                                          const float* __restrict__ lw,   // [64 x 32], rows 0..31 used
                                          const float* __restrict__ lb,   // [32]
                                          float* __restrict__ outA,
                                          float* __restrict__ outC) {
    const int m    = lane & 15;
    const int koff = (lane >> 4) ? 2 : 0;
#pragma unroll
    for (int nt = 0; nt < 2; ++nt) {
        v8f c = {};
#pragma unroll
        for (int kc = 0; kc < 8; ++kc) {
            const int k0 = kc * 4 + koff;
            v2f a, b;
            a.x = (m == 0) ? cw[k0]     : (m == 1) ? cb[k0]     : 0.0f;
            a.y = (m == 0) ? cw[k0 + 1] : (m == 1) ? cb[k0 + 1] : 0.0f;
            const int n = nt * 16 + m;
            b.x = lw[(k0)     * HIDDEN + n];
            b.y = lw[(k0 + 1) * HIDDEN + n];
            // D = A*B + C  (f32, exact) -> v_wmma_f32_16x16x4_f32
            c = __builtin_amdgcn_wmma_f32_16x16x4_f32(false, a, false, b,
                                                      (short)0, c, false, false);
        }
        if (lane < 16) {
            outA[nt * 16 + lane] = c[0];                       // row 0: conv_w @ W
            outC[nt * 16 + lane] = c[1] + lb[nt * 16 + lane];  // row 1: conv_b @ W + b
        }
    }
}

__global__ void k_precompute(const float* __restrict__ czw, const float* __restrict__ czb,
                             const float* __restrict__ lzw, const float* __restrict__ lzb,
                             const float* __restrict__ chw, const float* __restrict__ chb,
                             const float* __restrict__ lhw, const float* __restrict__ lhb,
                             const float* __restrict__ att,
                             float* __restrict__ cst /* az32 | cz32 | ah32 | ch32 | probs12 */) {
    const int lane = threadIdx.x;  // exactly one wave32, EXEC all-ones for WMMA
    gate_fuse(lane, czw, czb, lzw, lzb, cst + 0,  cst + 32);
    gate_fuse(lane, chw, chb, lhw, lhb, cst + 64, cst + 96);
    if (lane == 0) {               // softmax over the 12 attention logits
        float mx = att[0];
#pragma unroll
        for (int t = 1; t < PERIODS; ++t) mx = fmaxf(mx, att[t]);
        float e[PERIODS], s = 0.0f;
#pragma unroll
        for (int t = 0; t < PERIODS; ++t) { e[t] = __expf(att[t] - mx); s += e[t]; }
        const float inv = __builtin_amdgcn_rcpf(s);
#pragma unroll
        for (int t = 0; t < PERIODS; ++t) cst[128 + t] = e[t] * inv;
    }
}

// ---------------------------------------------------------------------------
// Kernel 1: weighted in-degree (deg already seeded with self-loop 1.0).
// ---------------------------------------------------------------------------
__global__ void k_degree(const long long* __restrict__ ei, const float* __restrict__ ew,
                         float* __restrict__ deg, int E) {
    int e = blockIdx.x * blockDim.x + threadIdx.x;
    if (e >= E) return;
    const int d = (int)ei[(size_t)E + e];   // dst row
    atomicAdd(&deg[d], ew[e]);
}

// Kernel 2: dinv = (deg>0 ? deg : 1)^-0.5
__global__ void k_dinv(const float* __restrict__ deg, float* __restrict__ dinv, int N) {
    int n = blockIdx.x * blockDim.x + threadIdx.x;
    if (n >= N) return;
    float dg = deg[n];
    dg = dg > 0.0f ? dg : 1.0f;
    dinv[n] = rsqrtf(dg);
}

// ---------------------------------------------------------------------------
// Kernel 3: edge scatter  agg[d, :] += dinv[s]*w*dinv[d] * x[s, :]
// x rows are 48B (16B aligned) -> three float4 loads. agg/x/dinv live in L2.
// ---------------------------------------------------------------------------
__global__ void k_scatter(const long long* __restrict__ ei, const float* __restrict__ ew,
                          const float* __restrict__ dinv, const float* __restrict__ x,
                          float* __restrict__ agg, int E) {
    int e = blockIdx.x * blockDim.x + threadIdx.x;
    if (e >= E) return;
    const int s = (int)ei[e];
    const int d = (int)ei[(size_t)E + e];
    const float w = dinv[s] * ew[e] * dinv[d];
    const float4* xr = (const float4*)(x + (size_t)s * PERIODS);
    const float4 x0 = xr[0], x1 = xr[1], x2 = xr[2];
    float* ad = agg + (size_t)d * PERIODS;
    atomicAdd(ad + 0,  w * x0.x); atomicAdd(ad + 1,  w * x0.y);
    atomicAdd(ad + 2,  w * x0.z); atomicAdd(ad + 3,  w * x0.w);
    atomicAdd(ad + 4,  w * x1.x); atomicAdd(ad + 5,  w * x1.y);
    atomicAdd(ad + 6,  w * x1.z); atomicAdd(ad + 7,  w * x1.w);
    atomicAdd(ad + 8,  w * x2.x); atomicAdd(ad + 9,  w * x2.y);
    atomicAdd(ad + 10, w * x2.z); atomicAdd(ad + 11, w * x2.w);
}

// ---------------------------------------------------------------------------
// Kernel 4: node epilogue. One wave32 per node, lane = hidden channel.
//   a[n,t] = agg[n,t] + dinv[n]^2 * x[n,t]          (self loop)
//   H_accum[k] = sum_t p[t] * (1-sigmoid(a*az+cz)) * tanh(a*ah+ch)
//   out[n] = relu(H_accum) . head_w + head_b
// ---------------------------------------------------------------------------
__device__ __forceinline__ float fast_sigmoid(float v) {
    return __builtin_amdgcn_rcpf(1.0f + __expf(-v));
}
__device__ __forceinline__ float fast_tanh(float v) {
    const float e2 = __expf(2.0f * v);
    return (e2 - 1.0f) * __builtin_amdgcn_rcpf(e2 + 1.0f);
}

__global__ __launch_bounds__(256) void k_node(const float* __restrict__ agg,
                                              const float* __restrict__ dinv,
                                              const float* __restrict__ x,
                                              const float* __restrict__ cst,
                                              const float* __restrict__ head_w,
                                              const float* __restrict__ head_b,
                                              float* __restrict__ out, int N) {
    __shared__ float sc[128 + PERIODS];
    for (int i = threadIdx.x; i < 128 + PERIODS; i += blockDim.x) sc[i] = cst[i];
    __syncthreads();

    const int lane = threadIdx.x & 31;
    const int wv   = threadIdx.x >> 5;
    const int n    = blockIdx.x * (int)(blockDim.x >> 5) + wv;
    if (n >= N) return;

    const float az = sc[lane], cz = sc[32 + lane];
    const float ah = sc[64 + lane], ch = sc[96 + lane];

    const float di = dinv[n];
    const float d2 = di * di;
    float a_l = 0.0f;
    if (lane < PERIODS)
        a_l = agg[(size_t)n * PERIODS + lane] + d2 * x[(size_t)n * PERIODS + lane];

    float acc = 0.0f;
#pragma unroll
    for (int t = 0; t < PERIODS; ++t) {
        const float a = __shfl(a_l, t, 32);       // broadcast agg scalar for period t
        const float p = sc[128 + t];
        const float z  = fast_sigmoid(a * az + cz);
        const float ht = fast_tanh(a * ah + ch);
        acc += p * (1.0f - z) * ht;               // Hc = (1-Z)*Ht  (Z*H term = 0)
    }

    float v = fmaxf(acc, 0.0f) * head_w[lane];    // relu then head dot
#pragma unroll
    for (int off = 16; off > 0; off >>= 1) v += __shfl_xor(v, off, 32);
    if (lane == 0) out[n] = v + head_b[0];
}

// ---------------------------------------------------------------------------
// Launch
// ---------------------------------------------------------------------------
extern "C" void kernel_launch(void* const* d_in, const int* in_sizes, int n_in,
                              void* d_out, int out_size, void* d_ws, size_t ws_size,
                              hipStream_t stream) {
    const float*     x      = (const float*)d_in[0];
    const long long* ei     = (const long long*)d_in[1];
    const float*     ew     = (const float*)d_in[2];
    const float*     att    = (const float*)d_in[3];
    const float*     czw    = (const float*)d_in[4];
    const float*     czb    = (const float*)d_in[5];
    const float*     lzw    = (const float*)d_in[6];
    const float*     lzb    = (const float*)d_in[7];
    // d_in[8..11] = r-gate params: dead code when H=0 (only used via H*R).
    const float*     chw    = (const float*)d_in[12];
    const float*     chb    = (const float*)d_in[13];
    const float*     lhw    = (const float*)d_in[14];
    const float*     lhb    = (const float*)d_in[15];
    const float*     head_w = (const float*)d_in[16];
    const float*     head_b = (const float*)d_in[17];

    const int N = in_sizes[0] / PERIODS;
    const int E = in_sizes[1] / 2;

    float* ws   = (float*)d_ws;
    float* deg  = ws;                       // N
    float* agg  = ws + (size_t)N;           // N*12
    float* dinv = ws + (size_t)13 * N;      // N
    float* cst  = ws + (size_t)14 * N;      // 140
    float* out  = (float*)d_out;

    const int B = 256;
    k_init<<<(N * PERIODS + B - 1) / B, B, 0, stream>>>(deg, agg, N);
    k_precompute<<<1, 32, 0, stream>>>(czw, czb, lzw, lzb, chw, chb, lhw, lhb, att, cst);
    k_degree<<<(E + B - 1) / B, B, 0, stream>>>(ei, ew, deg, E);
    k_dinv<<<(N + B - 1) / B, B, 0, stream>>>(deg, dinv, N);
    k_scatter<<<(E + B - 1) / B, B, 0, stream>>>(ei, ew, dinv, x, agg, E);
    k_node<<<(N + 7) / 8, B, 0, stream>>>(agg, dinv, x, cst, head_w, head_b, out, N);
}